// ourLoss_14104672600659
// MI455X (gfx1250) — compile-verified
//
#include <hip/hip_runtime.h>
#include <hip/hip_bf16.h>

typedef __attribute__((ext_vector_type(2))) float v2f;
typedef __attribute__((ext_vector_type(8))) float v8f;

// Stable log-sigmoid, matches jax.nn.log_sigmoid: -softplus(-x)
__device__ __forceinline__ float lsig(float x) {
    return fminf(x, 0.0f) - log1pf(expf(-fabsf(x)));
}

// Reduce 256 per-thread partials to a single f32 using the CDNA5 f32 matrix
// pipe: four chained V_WMMA_F32_16X16X4_F32 ops with an all-ones B matrix.
// Each WMMA consumes a 16x4 tile (64 values) of partials; D[m,n] accumulates
// row-sums identically across all n. Lane L (<16) holds D[0..7, L] in its 8
// accumulator VGPRs, lane L+16 holds D[8..15, L]; summing the 8 components and
// adding the xor-16 partner gives the full 256-value sum. Result valid on
// lane 0 of wave 0. Everything stays f32 (RNE), no precision loss.
__device__ __forceinline__ float blockReduceWmma(float v, float* lds) {
    lds[threadIdx.x] = v;
    __syncthreads();
    float total = 0.0f;
    if (threadIdx.x < 32) {              // wave 0 fully active -> EXEC all 1s
        const int lane = threadIdx.x;
        v2f bmat;
        bmat[0] = 1.0f;
        bmat[1] = 1.0f;                  // all-ones 4x16 B matrix
        v8f c = {};
#pragma unroll
        for (int chunk = 0; chunk < 4; ++chunk) {
            v2f a;
            a[0] = lds[chunk * 64 + lane * 2 + 0];
            a[1] = lds[chunk * 64 + lane * 2 + 1];
            // D = A(16x4, f32) x B(4x16, ones) + C   (accumulator chained)
            c = __builtin_amdgcn_wmma_f32_16x16x4_f32(
                    false, a, false, bmat, (short)0, c, false, false);
        }
        float s = c[0] + c[1] + c[2] + c[3] + c[4] + c[5] + c[6] + c[7];
        total = s + __shfl_xor(s, 16, 32);   // combine M=0..7 with M=8..15
    }
    return total;                            // valid on lane 0
}

// Pass 1: grid-stride over all 4*8*NH elements; gather b/h/th rows, apply the
// group-specific hinge, block-reduce via WMMA, one partial per block.
__global__ __launch_bounds__(256) void loss_partial_kernel(
    const float* __restrict__ logits,
    const int* __restrict__ cd_hits, const int* __restrict__ cd_bcol, const int* __restrict__ cd_hcol,
    const int* __restrict__ ci_hits, const int* __restrict__ ci_rev,
    const int* __restrict__ ci_bcol, const int* __restrict__ ci_hcol,
    const int* __restrict__ hd_hits, const int* __restrict__ hd_bcol,
    const int* __restrict__ hd_hcol, const float* __restrict__ hd_lb,
    const int* __restrict__ hi_hits, const int* __restrict__ hi_rev,
    const int* __restrict__ hi_bcol, const int* __restrict__ hi_hcol,
    int NH, int E, float* __restrict__ ws)
{
    constexpr int ROWSTRIDE = 97;   // R + 1
    __shared__ float lds[256];

    const int stride = gridDim.x * blockDim.x;
    float acc = 0.0f;

    for (int e = blockIdx.x * blockDim.x + threadIdx.x; e < E; e += stride) {
        const int g   = e / (8 * NH);
        const int rem = e - g * (8 * NH);
        const int r   = rem / NH;
        const int j   = rem - r * NH;

        const int* hits;
        const int* rev;
        const int* bcolp;
        const int* hcolp;
        float beta = 0.0f;
        bool  head;
        if (g == 0)      { hits = cd_hits; rev = nullptr; bcolp = cd_bcol; hcolp = cd_hcol; head = false; }
        else if (g == 1) { hits = ci_hits; rev = ci_rev;  bcolp = ci_bcol; hcolp = ci_hcol; head = false; }
        else if (g == 2) { hits = hd_hits; rev = nullptr; bcolp = hd_bcol; hcolp = hd_hcol; head = true; beta = hd_lb[r]; }
        else             { hits = hi_hits; rev = hi_rev;  bcolp = hi_bcol; hcolp = hi_hcol; head = true; }

        const int hidx = hits[r * NH + j];
        const int bidx = rev ? rev[r * NH + j] : hidx;

        const long hoff = (long)hidx * ROWSTRIDE;
        const long boff = (long)bidx * ROWSTRIDE;

        // three independent gathers (same-row locality for b/th)
        const float h  = logits[hoff + hcolp[r]];
        const float b  = logits[boff + bcolp[r]];
        const float th = logits[boff];

        const float lsb = lsig(b);
        const float lsh = lsig(h);

        float v;
        if (!head) v = (b > th)  ? fmaxf(lsb - lsh, 0.0f)        : 0.0f;  // cd / ci
        else       v = (b <= th) ? fmaxf(beta + lsh - lsb, 0.0f) : 0.0f;  // hd (beta) / hi (beta=0)
        acc += v;
    }

    const float total = blockReduceWmma(acc, lds);
    if (threadIdx.x == 0) ws[blockIdx.x] = total;
}

// Pass 2: deterministically fold block partials + Sum(logbeta) into d_out[0].
__global__ __launch_bounds__(256) void loss_final_kernel(
    const float* __restrict__ ws, int nblk,
    const float* __restrict__ cd_lb, const float* __restrict__ ci_lb,
    const float* __restrict__ hd_lb, const float* __restrict__ hi_lb,
    float* __restrict__ out)
{
    __shared__ float lds[256];
    float acc = 0.0f;
    for (int i = threadIdx.x; i < nblk; i += 256) acc += ws[i];
    const float total = blockReduceWmma(acc, lds);
    if (threadIdx.x == 0) {
        float lb = 0.0f;
#pragma unroll
        for (int r = 0; r < 8; ++r)
            lb += cd_lb[r] + ci_lb[r] + hd_lb[r] + hi_lb[r];
        out[0] = total + lb;
    }
}

extern "C" void kernel_launch(void* const* d_in, const int* in_sizes, int n_in,
                              void* d_out, int out_size, void* d_ws, size_t ws_size,
                              hipStream_t stream) {
    const float* logits  = (const float*)d_in[0];
    const int*   cd_hits = (const int*)d_in[1];
    const int*   cd_bcol = (const int*)d_in[2];
    const int*   cd_hcol = (const int*)d_in[3];
    const float* cd_lb   = (const float*)d_in[4];
    const int*   ci_hits = (const int*)d_in[5];
    const int*   ci_rev  = (const int*)d_in[6];
    const int*   ci_bcol = (const int*)d_in[7];
    const int*   ci_hcol = (const int*)d_in[8];
    const float* ci_lb   = (const float*)d_in[9];
    const int*   hd_hits = (const int*)d_in[10];
    const int*   hd_bcol = (const int*)d_in[11];
    const int*   hd_hcol = (const int*)d_in[12];
    const float* hd_lb   = (const float*)d_in[13];
    const int*   hi_hits = (const int*)d_in[14];
    const int*   hi_rev  = (const int*)d_in[15];
    const int*   hi_bcol = (const int*)d_in[16];
    const int*   hi_hcol = (const int*)d_in[17];
    const float* hi_lb   = (const float*)d_in[18];

    const int NH = in_sizes[1] / 8;           // hits per row (12800)
    const int E  = 4 * 8 * NH;                // total elements (409600)

    int nblk = (E + 255) / 256;               // 1600 blocks -> 12.8K wave32s
    if (nblk > 2048) nblk = 2048;             // keep ws usage <= 8 KB

    float* ws = (float*)d_ws;

    loss_partial_kernel<<<nblk, 256, 0, stream>>>(
        logits,
        cd_hits, cd_bcol, cd_hcol,
        ci_hits, ci_rev, ci_bcol, ci_hcol,
        hd_hits, hd_bcol, hd_hcol, hd_lb,
        hi_hits, hi_rev, hi_bcol, hi_hcol,
        NH, E, ws);

    loss_final_kernel<<<1, 256, 0, stream>>>(
        ws, nblk, cd_lb, ci_lb, hd_lb, hi_lb, (float*)d_out);
}